// MLP_33904471835187
// MI455X (gfx1250) — compile-verified
//
#include <hip/hip_runtime.h>

typedef __attribute__((ext_vector_type(2))) float v2f;
typedef __attribute__((ext_vector_type(8))) float v8f;

#define B_SZ   4
#define DEC_N  128
#define ENC_N  512
#define HID_N  512
#define K_N    512

// C[M,N] = A[M,K] @ W[N, ldw]^T (using columns [woff, woff+K) of W) + optional bias[n]
// One wave per 32x32 output block = 2x2 WMMA tiles (fp32 16x16x4).
// Per k-step: 4 x b64 fragment loads -> 4 WMMAs (1:1 load:wmma ratio, and four
// independent accumulator chains to hide WMMA latency).
// 8 waves (256 threads) per block; grids sized so unit count divides exactly
// (no divergence -> EXEC is all 1s around every WMMA, as the ISA requires).
__global__ __launch_bounds__(256) void proj_wmma_f32(
    const float* __restrict__ A, const float* __restrict__ W,
    const float* __restrict__ bias, float* __restrict__ C,
    int N, int K, int ldw, int woff, int nunits_n)
{
    const int lane = threadIdx.x & 31;
    const int wave = threadIdx.x >> 5;
    const int unit = blockIdx.x * 8 + wave;
    const int um   = (unit / nunits_n) << 5;   // 32-row block base
    const int un   = (unit % nunits_n) << 5;   // 32-col block base

    const int half = lane >> 4;      // 0 or 1
    const int r    = lane & 15;      // row-in-tile (M for A frags, N for B frags)

    // A frags: lane holds A[row][k + 2*half + {0,1}] -> b64 loads
    const float* a0p = A + (size_t)(um + r)      * K + (half << 1);
    const float* a1p = A + (size_t)(um + 16 + r) * K + (half << 1);
    // B frags: lane holds W[col][woff + k + 2*half + {0,1}] -> b64 loads
    const float* b0p = W + (size_t)(un + r)      * ldw + woff + (half << 1);
    const float* b1p = W + (size_t)(un + 16 + r) * ldw + woff + (half << 1);

    v8f c00 = {0.f,0.f,0.f,0.f,0.f,0.f,0.f,0.f};
    v8f c01 = c00, c10 = c00, c11 = c00;

    for (int k = 0; k < K; k += 4) {
        v2f a0 = *(const v2f*)(a0p + k);
        v2f a1 = *(const v2f*)(a1p + k);
        v2f b0 = *(const v2f*)(b0p + k);
        v2f b1 = *(const v2f*)(b1p + k);
        // 8 args: (neg_a, A, neg_b, B, c_mod, C, reuse_a, reuse_b)
        c00 = __builtin_amdgcn_wmma_f32_16x16x4_f32(false, a0, false, b0, (short)0, c00, false, false);
        c01 = __builtin_amdgcn_wmma_f32_16x16x4_f32(false, a0, false, b1, (short)0, c01, false, false);
        c10 = __builtin_amdgcn_wmma_f32_16x16x4_f32(false, a1, false, b0, (short)0, c10, false, false);
        c11 = __builtin_amdgcn_wmma_f32_16x16x4_f32(false, a1, false, b1, (short)0, c11, false, false);
    }

    // C/D layout: vgpr v -> m = v + 8*half, n = r
    const int   n0   = un + r;
    const int   n1   = un + 16 + r;
    const float bad0 = bias ? bias[n0] : 0.f;
    const float bad1 = bias ? bias[n1] : 0.f;
    float* row0 = C + (size_t)(um + (half << 3)) * N;        // M tile 0
    float* row1 = C + (size_t)(um + 16 + (half << 3)) * N;   // M tile 1
#pragma unroll
    for (int v = 0; v < 8; ++v) {
        row0[(size_t)v * N + n0] = c00[v] + bad0;
        row0[(size_t)v * N + n1] = c01[v] + bad1;
        row1[(size_t)v * N + n0] = c10[v] + bad0;
        row1[(size_t)v * N + n1] = c11[v] + bad1;
    }
}

// attn[b,q,e] = b_out + sum_h relu(dec_proj[b,q,h] + enc_proj[b,e,h]) * w_out[h]
// One block per (b,q); dec row (b_mlp folded in) and w_out staged in LDS;
// one wave per e, lanes split HID with coalesced float4 loads of enc_proj,
// then a wave32 shfl_xor reduction.
__global__ __launch_bounds__(256) void relu_dot_score(
    const float* __restrict__ dec_proj,   // [B*DEC, HID] (bias folded in)
    const float* __restrict__ enc_proj,   // [B*ENC, HID]
    const float* __restrict__ w_out,      // [HID]
    const float* __restrict__ b_out,      // [1]
    float* __restrict__ out)              // [B*DEC, ENC]
{
    __shared__ float s_dp[HID_N];
    __shared__ float s_w [HID_N];

    const int bq  = blockIdx.x;           // 0 .. B*DEC-1
    const int b   = bq >> 7;              // DEC = 128
    const int tid = threadIdx.x;

    s_dp[tid]       = dec_proj[(size_t)bq * HID_N + tid];
    s_dp[tid + 256] = dec_proj[(size_t)bq * HID_N + tid + 256];
    s_w [tid]       = w_out[tid];
    s_w [tid + 256] = w_out[tid + 256];
    __syncthreads();

    const int   lane = tid & 31;
    const int   wave = tid >> 5;
    const float bo   = b_out[0];
    const float* ep_base = enc_proj + (size_t)b * ENC_N * HID_N;

    for (int e = wave; e < ENC_N; e += 8) {
        const float* ep = ep_base + (size_t)e * HID_N;
        float acc = 0.f;
#pragma unroll
        for (int j = 0; j < 4; ++j) {
            const int h = (lane << 2) + (j << 7);   // coalesced across lanes
            float4 ev = *(const float4*)(ep + h);
            float4 dv = *(const float4*)(s_dp + h);
            float4 wv = *(const float4*)(s_w + h);
            float t0 = fmaxf(dv.x + ev.x, 0.f);
            float t1 = fmaxf(dv.y + ev.y, 0.f);
            float t2 = fmaxf(dv.z + ev.z, 0.f);
            float t3 = fmaxf(dv.w + ev.w, 0.f);
            acc = fmaf(t0, wv.x, acc);
            acc = fmaf(t1, wv.y, acc);
            acc = fmaf(t2, wv.z, acc);
            acc = fmaf(t3, wv.w, acc);
        }
        // wave32 reduction
        for (int off = 16; off > 0; off >>= 1)
            acc += __shfl_xor(acc, off, 32);
        if (lane == 0)
            out[(size_t)bq * ENC_N + e] = acc + bo;
    }
}

extern "C" void kernel_launch(void* const* d_in, const int* in_sizes, int n_in,
                              void* d_out, int out_size, void* d_ws, size_t ws_size,
                              hipStream_t stream) {
    const float* dec_states = (const float*)d_in[0]; // (4,128,512)
    const float* enc_states = (const float*)d_in[1]; // (4,512,512)
    const float* W_mlp      = (const float*)d_in[2]; // (512, 1024) row-major
    const float* b_mlp      = (const float*)d_in[3]; // (512,)
    const float* W_out      = (const float*)d_in[4]; // (1,512)
    const float* b_out      = (const float*)d_in[5]; // (1,)
    float*       out        = (float*)d_out;         // (4,128,512)

    float* enc_proj = (float*)d_ws;                            // 2048*512 f32 = 4 MB
    float* dec_proj = enc_proj + (size_t)B_SZ * ENC_N * HID_N; // 512*512 f32 = 1 MB

    // enc_proj: M = 2048 -> 64 x 16 = 1024 wave-units -> 128 blocks of 8 waves
    proj_wmma_f32<<<128, 256, 0, stream>>>(enc_states, W_mlp, nullptr, enc_proj,
                                           HID_N, K_N,
                                           /*ldw=*/1024, /*woff=*/0,
                                           /*nunits_n=*/16);
    // dec_proj: M = 512 -> 16 x 16 = 256 wave-units -> 32 blocks; fold b_mlp
    proj_wmma_f32<<<32, 256, 0, stream>>>(dec_states, W_mlp, b_mlp, dec_proj,
                                          HID_N, K_N,
                                          /*ldw=*/1024, /*woff=*/512,
                                          /*nunits_n=*/16);
    // score: one block per (b,q) row
    relu_dot_score<<<B_SZ * DEC_N, 256, 0, stream>>>(dec_proj, enc_proj,
                                                     W_out, b_out, out);
}